// SPADHistNV_39290360823831
// MI455X (gfx1250) — compile-verified
//
#include <hip/hip_runtime.h>
#include <math.h>

typedef __attribute__((ext_vector_type(2))) float v2f;
typedef __attribute__((ext_vector_type(8))) float v8f;

#define NBINS     64
#define RR        65536      // R*R pixels per camera
#define PIX_BLOCK 8192       // pixels per block (8 waves * 1024)
#define PIX_WAVE  1024       // pixels per wave (32 lanes * 32 iters)
#define ITERS     32

// Single tanh; v_nop satisfies the TRANS-op hazard (1 independent op or V_NOP
// after a trans op before its output is consumed).
__device__ __forceinline__ float tanh1(float x) {
    float t;
    asm("v_tanh_f32 %0, %1\n\tv_nop" : "=v"(t) : "v"(x));
    return t;
}

// Four independent tanh in one block: each covers the previous one's TRANS
// hazard; one trailing v_nop protects the last. Early-clobber outputs so the
// register allocator cannot alias an output onto a later tanh's input.
__device__ __forceinline__ void tanh4(float x0, float x1, float x2, float x3,
                                      float& t0, float& t1, float& t2, float& t3) {
    float a, b, c, d;
    asm("v_tanh_f32 %0, %4\n\t"
        "v_tanh_f32 %1, %5\n\t"
        "v_tanh_f32 %2, %6\n\t"
        "v_tanh_f32 %3, %7\n\t"
        "v_nop"
        : "=&v"(a), "=&v"(b), "=&v"(c), "=&v"(d)
        : "v"(x0), "v"(x1), "v"(x2), "v"(x3));
    t0 = a; t1 = b; t2 = c; t3 = d;
}

__global__ void __launch_bounds__(256)
spad_zero(float* out, int n) {
    int i = blockIdx.x * 256 + threadIdx.x;
    if (i < n) out[i] = 0.0f;
}

__global__ void __launch_bounds__(256)
spad_hist(const float* __restrict__ normals,
          const float* __restrict__ inters,
          const float* __restrict__ film,
          const float* __restrict__ cosm,
          float* __restrict__ out,
          float invBin, float inv2sig2, float radScale)
{
    __shared__ float lds[8][32][NBINS];   // 64 KB of the 320 KB WGP LDS

    const int lane = threadIdx.x & 31;
    const int wave = threadIdx.x >> 5;
    const int base = blockIdx.x * PIX_BLOCK + wave * PIX_WAVE;  // global pixel id
    const int cam  = blockIdx.x >> 3;     // 8 blocks per camera

    float acc[NBINS];
#pragma unroll
    for (int k = 0; k < NBINS; ++k) acc[k] = 0.0f;

    for (int it = 0; it < ITERS; ++it) {
        const int p  = base + it * 32 + lane;      // coalesced: lane = pixel
        const int pm = p & (RR - 1);               // pixel within image

        const float nx = normals[p * 3 + 0];
        const float ny = normals[p * 3 + 1];
        const float nz = normals[p * 3 + 2];
        const float ix = inters[p * 3 + 0];
        const float iy = inters[p * 3 + 1];
        const float iz = inters[p * 3 + 2];
        const float fx = film[pm * 3 + 0];
        const float fy = film[pm * 3 + 1];
        const float fz = film[pm * 3 + 2];
        const float cm = cosm[pm];

        // cosine foreshortening, clipped
        float dt = fx * nx + fy * ny + fz * nz;
        dt = fminf(fmaxf(dt, 0.0f), 1.0f);

        // laser geometry: origin offset 2*DCS along +x
        const float lx = ix - 0.002f;
        const float ly = iy;
        const float lz = iz;
        const float ld2 = lx * lx + ly * ly + lz * lz;
        const float ld  = sqrtf(ld2);
        const float tan2 = (lx * lx + ly * ly) / (lz * lz);
        const float lmask = __expf(-tan2 * inv2sig2);

        const float r  = dt * lmask * cm * cm * cm * radScale / ld2;
        const float rh = 0.5f * r;   // sigma_k - sigma_{k+1} = 0.5*(t_k - t_{k+1})

        const float depth = 0.5f * (sqrtf(ix * ix + iy * iy + iz * iz) + ld);
        const float u = depth * invBin;            // d_bin
        const float h = 1.5f * u;                  // x_k = h - 1.5*k

        // soft histogram on raw tanh values: t_k = tanh(1.5*(u-k))
        float tp = tanh1(h);                       // t at k = 0
#pragma unroll
        for (int k = 0; k < NBINS; k += 4) {
            float t1, t2, t3, t4;
            tanh4(h - 1.5f * (float)(k + 1),
                  h - 1.5f * (float)(k + 2),
                  h - 1.5f * (float)(k + 3),
                  h - 1.5f * (float)(k + 4),
                  t1, t2, t3, t4);
            acc[k + 0] = __builtin_fmaf(rh, tp - t1, acc[k + 0]);
            acc[k + 1] = __builtin_fmaf(rh, t1 - t2, acc[k + 1]);
            acc[k + 2] = __builtin_fmaf(rh, t2 - t3, acc[k + 2]);
            acc[k + 3] = __builtin_fmaf(rh, t3 - t4, acc[k + 3]);
            tp = t4;
        }
    }

    // --- cross-lane reduction of per-lane 64-bin histograms via WMMA ---
    // Stage histograms to LDS so they can be reloaded in B-matrix layout.
#pragma unroll
    for (int k = 0; k < NBINS; ++k) lds[wave][lane][k] = acc[k];
    __syncthreads();

    // B (4x16 f32) layout: VGPR0 = rows {0 | 2}, VGPR1 = rows {1 | 3};
    // lanes 0-15 carry columns n for rows {0,1}, lanes 16-31 for rows {2,3}.
    const int half = lane >> 4;      // which row-pair this lane supplies
    const int ncol = lane & 15;      // column (bin within 16-bin group)
    const v2f a = {1.0f, 1.0f};      // all-ones A => D[m,n] = sum_k B[k,n]

    float res[4];
#pragma unroll
    for (int grp = 0; grp < 4; ++grp) {
        v8f d = {};
#pragma unroll
        for (int i = 0; i < 8; ++i) {            // 8 x (4 lanes) = 32 lanes
            v2f b;
            b.x = lds[wave][i * 4 + 2 * half + 0][grp * 16 + ncol];
            b.y = lds[wave][i * 4 + 2 * half + 1][grp * 16 + ncol];
            d = __builtin_amdgcn_wmma_f32_16x16x4_f32(
                    false, a, false, b, (short)0, d, false, false);
        }
        // All D rows identical; VGPR0 lanes 0-15 hold bins grp*16 + 0..15.
        res[grp] = d[0];
    }

    // Divergent region only after the last WMMA (EXEC must be all-1s for WMMA).
    if (lane < 16) {
#pragma unroll
        for (int grp = 0; grp < 4; ++grp)
            atomicAdd(out + cam * NBINS + grp * 16 + lane, res[grp]);
    }
}

extern "C" void kernel_launch(void* const* d_in, const int* in_sizes, int n_in,
                              void* d_out, int out_size, void* d_ws, size_t ws_size,
                              hipStream_t stream) {
    const float* normals = (const float*)d_in[0];
    const float* inters  = (const float*)d_in[1];
    const float* film    = (const float*)d_in[2];
    const float* cosm    = (const float*)d_in[3];
    float* out = (float*)d_out;

    // Host-side scalar constants (match reference exactly)
    const double fov_half = 33.0 * M_PI / 360.0;
    const double width    = 2.0 * tan(fov_half);
    const double sig      = tan(21.5 * M_PI / 180.0) / 1.4;
    const float invBin    = (float)(1.0 / 0.0136);
    const float inv2sig2  = (float)(1.0 / (2.0 * sig * sig));
    const float radScale  = (float)(width * width / (M_PI * (double)RR));

    spad_zero<<<(out_size + 255) / 256, 256, 0, stream>>>(out, out_size);

    const int total_pixels = 32 * RR;               // BC * R * R
    const int blocks = total_pixels / PIX_BLOCK;    // 256
    spad_hist<<<blocks, 256, 0, stream>>>(normals, inters, film, cosm, out,
                                          invBin, inv2sig2, radScale);
}